// GCNClassifier_11467562680667
// MI455X (gfx1250) — compile-verified
//
#include <hip/hip_runtime.h>

// ---------------------------------------------------------------------------
// GCN classifier for MI455X (gfx1250, wave32).
// GEMM uses V_WMMA_F32_16X16X4_F32 (fp32 matrix path, matches fp32 reference).
// All intermediates (~55 MB) are L2-resident (192 MB L2), so the multi-kernel
// pipeline pays HBM only for the initial x read and the 16 KB output.
// ---------------------------------------------------------------------------

typedef float v2f __attribute__((ext_vector_type(2)));
typedef float v8f __attribute__((ext_vector_type(8)));

#define DCH 64            // channels (in == hidden)
#define NGRAPH 64

// ---------------------------------------------------------------------------
// Utility: zero-fill
// ---------------------------------------------------------------------------
__global__ void k_zero(float* __restrict__ p, int n) {
    int i = blockIdx.x * blockDim.x + threadIdx.x;
    if (i < n) p[i] = 0.0f;
}

// Degree accumulation: deg[dst[e]] += 1  (deg buffer pre-zeroed)
__global__ void k_deg(const int* __restrict__ dst, float* __restrict__ deg, int nE) {
    int e = blockIdx.x * blockDim.x + threadIdx.x;
    if (e < nE) unsafeAtomicAdd(&deg[dst[e]], 1.0f);
}

// dinv[n] = rsqrt(deg[n] + 1)   (in place over the deg buffer)
__global__ void k_dinv(float* __restrict__ deg, int n) {
    int i = blockIdx.x * blockDim.x + threadIdx.x;
    if (i < n) deg[i] = rsqrtf(deg[i] + 1.0f);
}

// coef[e] = dinv[src[e]] * dinv[dst[e]]
__global__ void k_coef(const int* __restrict__ src, const int* __restrict__ dst,
                       const float* __restrict__ dinv, float* __restrict__ coef, int nE) {
    int e = blockIdx.x * blockDim.x + threadIdx.x;
    if (e < nE) coef[e] = dinv[src[e]] * dinv[dst[e]];
}

// ---------------------------------------------------------------------------
// XW = X @ W via V_WMMA_F32_16X16X4_F32.
// One wave per 16-col tile; B fragments (16 k-steps x v2f) preloaded once in
// registers; row tiles streamed.  Epilogue also writes
//   ACC = XW * dinv^2 + bias   (self-loop + bias init for the scatter phase).
// nrows must be a multiple of 16 (50000 = 3125*16).
// ---------------------------------------------------------------------------
__global__ __launch_bounds__(256)
void k_gemm_xw(const float* __restrict__ X, const float* __restrict__ W,
               const float* __restrict__ dinv, const float* __restrict__ bias,
               float* __restrict__ XW, float* __restrict__ ACC, int nrows) {
    const int lane = threadIdx.x & 31;
    const int wavesPerBlk = blockDim.x >> 5;
    const int gw = blockIdx.x * wavesPerBlk + (threadIdx.x >> 5);
    const int col0 = (gw & 3) * 16;            // 4 column tiles cover DCH=64
    const int rt0 = gw >> 2;
    const int rtStride = (gridDim.x * wavesPerBlk) >> 2;
    const int nRowTiles = nrows >> 4;

    const int half = lane >> 4;                // 0 | 1
    const int l16  = lane & 15;

    // --- Preload B (4x16 fp32 fragments): VGPR idx = K%2, lane-half = K/2 ---
    v2f b[16];
#pragma unroll
    for (int kk = 0; kk < 16; ++kk) {
        const int krow = kk * 4 + 2 * half;
        b[kk].x = W[(krow + 0) * DCH + col0 + l16];
        b[kk].y = W[(krow + 1) * DCH + col0 + l16];
    }
    const float bval = bias[col0 + l16];

    for (int rt = rt0; rt < nRowTiles; rt += rtStride) {
        const int row0 = rt * 16;
        const int arow = row0 + l16;           // A: lanes 0-15 rows, halves split K
        v8f c = {};
#pragma unroll
        for (int kk = 0; kk < 16; ++kk) {
            const float2 av =
                *(const float2*)(X + (long long)arow * DCH + kk * 4 + 2 * half);
            v2f a; a.x = av.x; a.y = av.y;
            c = __builtin_amdgcn_wmma_f32_16x16x4_f32(
                    /*neg_a=*/false, a, /*neg_b=*/false, b[kk],
                    /*c_mod=*/(short)0, c, /*reuse_a=*/false, /*reuse_b=*/false);
        }
        // --- Epilogue: C/D layout row = v + 8*half, col = l16 ---
#pragma unroll
        for (int v = 0; v < 8; ++v) {
            const int row = row0 + v + 8 * half;
            const long long idx = (long long)row * DCH + col0 + l16;
            const float xwv = c[v];
            XW[idx] = xwv;
            const float di = dinv[row];
            ACC[idx] = xwv * di * di + bval;   // self-loop term + bias
        }
    }
}

// ---------------------------------------------------------------------------
// Edge scatter: ACC[dst] += XW[src] * coef.  One thread per (edge, 4 channels):
// float4 gather from L2-resident XW + 4 native fp32 L2 atomics.
// ---------------------------------------------------------------------------
__global__ void k_edge_scatter(const int* __restrict__ src, const int* __restrict__ dst,
                               const float* __restrict__ coef,
                               const float* __restrict__ XW, float* __restrict__ ACC,
                               int nE) {
    const long long t = (long long)blockIdx.x * blockDim.x + threadIdx.x;
    const int e = (int)(t >> 4);
    if (e >= nE) return;
    const int c4 = (int)(t & 15) * 4;
    const int s = src[e], d = dst[e];
    const float cf = coef[e];
    const float4 xv = *(const float4*)(XW + (long long)s * DCH + c4);
    float* ap = ACC + (long long)d * DCH + c4;
    unsafeAtomicAdd(ap + 0, xv.x * cf);
    unsafeAtomicAdd(ap + 1, xv.y * cf);
    unsafeAtomicAdd(ap + 2, xv.z * cf);
    unsafeAtomicAdd(ap + 3, xv.w * cf);
}

// ---------------------------------------------------------------------------
// Activation: XOUT = leaky_relu(ACC [+ residual]), slope 0.01
// ---------------------------------------------------------------------------
__global__ void k_act(const float* __restrict__ ACC, const float* __restrict__ XRES,
                      float* __restrict__ XOUT, int n, int useRes) {
    int i = blockIdx.x * blockDim.x + threadIdx.x;
    if (i >= n) return;
    float v = ACC[i];
    if (useRes) v += XRES[i];
    XOUT[i] = v > 0.0f ? v : 0.01f * v;
}

// ---------------------------------------------------------------------------
// Global mean pool
// ---------------------------------------------------------------------------
__global__ void k_pool_accum(const float* __restrict__ X, const int* __restrict__ batch,
                             float* __restrict__ psum, int nnodes) {
    int t = blockIdx.x * blockDim.x + threadIdx.x;
    if (t >= nnodes * DCH) return;
    const int nidx = t >> 6;       // / DCH
    const int c = t & 63;
    unsafeAtomicAdd(&psum[batch[nidx] * DCH + c], X[t]);
}

__global__ void k_pool_count(const int* __restrict__ batch, float* __restrict__ pcnt,
                             int nnodes) {
    int i = blockIdx.x * blockDim.x + threadIdx.x;
    if (i < nnodes) unsafeAtomicAdd(&pcnt[batch[i]], 1.0f);
}

__global__ void k_pool_fin(const float* __restrict__ psum, const float* __restrict__ pcnt,
                           float* __restrict__ out) {
    int t = blockIdx.x * blockDim.x + threadIdx.x;
    if (t < NGRAPH * DCH) out[t] = psum[t] / fmaxf(pcnt[t >> 6], 1.0f);
}

// ---------------------------------------------------------------------------
// Host-side orchestration
// ---------------------------------------------------------------------------
extern "C" void kernel_launch(void* const* d_in, const int* in_sizes, int n_in,
                              void* d_out, int out_size, void* d_ws, size_t ws_size,
                              hipStream_t stream) {
    (void)n_in; (void)out_size; (void)ws_size;

    const float* x0   = (const float*)d_in[0];          // [N, 64]
    const int*   ei   = (const int*)d_in[1];            // [2, E]
    const int*   batch = (const int*)d_in[2];           // [N]
    const float* Ws   = (const float*)d_in[3];          // [3, 64, 64]
    const float* bs   = (const float*)d_in[4];          // [3, 64]
    float*       out  = (float*)d_out;                  // [64, 64]

    const int N = in_sizes[0] / DCH;                    // 50000
    const int E = in_sizes[1] / 2;                      // 800000
    const int nLayers = in_sizes[3] / (DCH * DCH);      // 3
    const int ND = N * DCH;

    const int* src = ei;
    const int* dst = ei + E;

    // --- workspace layout (floats), 256B-aligned chunks ---
    auto alignUp = [](long long v) { return (v + 63) & ~63LL; };
    float* base = (float*)d_ws;
    long long off = 0;
    float* dinv = base + off; off += alignUp(N);
    float* coef = base + off; off += alignUp(E);
    float* xw   = base + off; off += alignUp(ND);
    float* acc  = base + off; off += alignUp(ND);
    float* xb0  = base + off; off += alignUp(ND);
    float* xb1  = base + off; off += alignUp(ND);
    float* psum = base + off; off += alignUp(NGRAPH * DCH);
    float* pcnt = base + off; off += alignUp(NGRAPH);

    const int B = 256;
    auto g = [&](long long n) { return (int)((n + B - 1) / B); };

    // --- degrees, dinv, per-edge coefficients (once) ---
    k_zero<<<g(N), B, 0, stream>>>(dinv, N);
    k_deg<<<g(E), B, 0, stream>>>(dst, dinv, E);
    k_dinv<<<g(N), B, 0, stream>>>(dinv, N);
    k_coef<<<g(E), B, 0, stream>>>(src, dst, dinv, coef, E);

    // --- 3 GCN layers ---
    const float* xin = x0;
    float* xout = xb0;
    for (int l = 0; l < nLayers; ++l) {
        // GEMM: 125 blocks * 8 waves = 1000 waves; 4 col tiles, rows strided
        k_gemm_xw<<<125, 256, 0, stream>>>(xin, Ws + (long long)l * DCH * DCH,
                                           dinv, bs + (long long)l * DCH,
                                           xw, acc, N);
        k_edge_scatter<<<g((long long)E * 16), B, 0, stream>>>(src, dst, coef,
                                                               xw, acc, E);
        k_act<<<g(ND), B, 0, stream>>>(acc, xin, xout, ND, l > 0 ? 1 : 0);
        xin = xout;
        xout = (xout == xb0) ? xb1 : xb0;
    }

    // --- global mean pool ---
    k_zero<<<g(NGRAPH * DCH + NGRAPH), B, 0, stream>>>(psum, NGRAPH * DCH + NGRAPH);
    // pcnt is contiguous after psum in the zeroed range
    (void)pcnt;
    k_pool_accum<<<g((long long)ND), B, 0, stream>>>(xin, batch, psum, N);
    k_pool_count<<<g(N), B, 0, stream>>>(batch, psum + NGRAPH * DCH, N);
    k_pool_fin<<<g(NGRAPH * DCH), B, 0, stream>>>(psum, psum + NGRAPH * DCH, out);
}